// HGTEncoder_33208687133238
// MI455X (gfx1250) — compile-verified
//
#include <hip/hip_runtime.h>

// CDNA5 / gfx1250, wave32. All dense matmuls use V_WMMA_F32_16X16X4_F32 (full
// fp32 fidelity; total dense FLOPs ~15 GF is trivially small next to the
// ~3-4 GB edge gather/scatter traffic, so f32 WMMA costs nothing on the
// 23.3 TB/s HBM roofline while exercising the CDNA5 matrix pipe).
typedef __attribute__((ext_vector_type(2))) float v2f;
typedef __attribute__((ext_vector_type(8))) float v8f;

#define HGT_H 2   // heads
#define HGT_D 16  // per-head dim

__device__ __forceinline__ float gelu_exact(float x) {
    return 0.5f * x * (1.0f + erff(x * 0.7071067811865475f));
}

// monotonic float <-> uint mapping for atomicMax on floats
__device__ __forceinline__ unsigned f2key(float f) {
    unsigned u = __float_as_uint(f);
    return (u & 0x80000000u) ? ~u : (u | 0x80000000u);
}
__device__ __forceinline__ float key2f(unsigned k) {
    unsigned u = (k & 0x80000000u) ? (k & 0x7FFFFFFFu) : ~k;
    return __uint_as_float(u);
}

// C[N, NT*16] = act( preA(A[N,K]) @ W[K, NT*16] + bias )  (+ gated skip)
// One wave handles MT 16-row tiles x NT*16 cols; K-loop in steps of 4 via
// v_wmma_f32_16x16x4_f32. A frag: lane m=lane&15 holds row m, VGPR0/1 hold
// K = k+2*(lane>>4)+{0,1}. B frag mirrors it. C/D: VGPR j -> M = j + 8*(lane>>4).
// Branch-free inner loop: row addresses are clamped (no per-lane EXEC churn);
// out-of-range tiles are skipped wave-uniformly at the store stage.
// Requires N % 16 == 0 for full-tile stores (true here: 100000/200000/300000).
template <int NT, int MT>
__global__ void gemm_bias_act(const float* __restrict__ A, int lda,
                              const float* __restrict__ W,     // [K, NT*16] row-major
                              const float* __restrict__ bias,  // [NT*16] or null
                              float* __restrict__ C, int ldc,
                              int N, int K,
                              const float* __restrict__ denom, // [N,2] or null: A <- gelu(A/(denom+eps))
                              const float* __restrict__ skipX, int ldskip,
                              const float* __restrict__ skipGate, // scalar or null
                              int reluOut)
{
    const int lane = threadIdx.x;       // 0..31
    const int m    = lane & 15;
    const int kh   = lane >> 4;         // 0..1
    const int row0 = (blockIdx.x * blockDim.y + threadIdx.y) * (16 * MT);
    if (row0 >= N) return;              // wave-uniform

    const int dout = NT * 16;

    // clamped per-tile A base pointers (branch-free; garbage rows never stored)
    const float* aBase[MT];
    int rowc[MT];
#pragma unroll
    for (int mt = 0; mt < MT; ++mt) {
        int r = row0 + mt * 16 + m;
        rowc[mt] = (r < N) ? r : (N - 1);
        aBase[mt] = A + (size_t)rowc[mt] * lda + 2 * kh;
    }

    v8f acc[MT][NT] = {};

    for (int k = 0; k < K; k += 4) {
        v2f av[MT];
#pragma unroll
        for (int mt = 0; mt < MT; ++mt) {
            const float* ap = aBase[mt] + k;
            float a0 = ap[0], a1 = ap[1];
            if (denom != nullptr) {  // segment-softmax normalize + exact GELU
                float dnm = denom[(size_t)rowc[mt] * 2 + ((k >> 4) & 1)] + 1e-16f;
                a0 = gelu_exact(a0 / dnm);
                a1 = gelu_exact(a1 / dnm);
            }
            av[mt].x = a0; av[mt].y = a1;
        }
        const float* wp = W + (size_t)(k + 2 * kh) * dout + m;
        v2f bv[NT];
#pragma unroll
        for (int t = 0; t < NT; ++t) {
            bv[t].x = wp[t * 16];
            bv[t].y = wp[dout + t * 16];
        }
#pragma unroll
        for (int mt = 0; mt < MT; ++mt)
#pragma unroll
            for (int t = 0; t < NT; ++t)
                acc[mt][t] = __builtin_amdgcn_wmma_f32_16x16x4_f32(
                    false, av[mt], false, bv[t], (short)0, acc[mt][t], false, false);
    }

    float sk = 1.0f, skc = 0.0f;
    if (skipGate != nullptr) {
        sk = 1.0f / (1.0f + expf(-skipGate[0]));
        skc = 1.0f - sk;
    }

#pragma unroll
    for (int mt = 0; mt < MT; ++mt) {
        if (row0 + mt * 16 >= N) continue;  // wave-uniform tile guard
#pragma unroll
        for (int t = 0; t < NT; ++t) {
#pragma unroll
            for (int j = 0; j < 8; ++j) {
                const int row = row0 + mt * 16 + j + 8 * kh;
                const int col = t * 16 + m;
                float v = acc[mt][t][j];
                if (bias != nullptr) v += bias[col];
                if (skipGate != nullptr) v = sk * v + skc * skipX[(size_t)row * ldskip + col];
                if (reluOut) v = fmaxf(v, 0.0f);
                C[(size_t)row * ldc + col] = v;
            }
        }
    }
}

// out[n, h*16+e] = sum_d IN[n, colOff + h*16 + d] * WREL[h, d, e]   (K=16 WMMA)
// Requires N % 16 == 0 (true here).
__global__ void rel_transform(const float* __restrict__ IN, int ldin, int colOff,
                              const float* __restrict__ WREL,  // [H,16,16]
                              float* __restrict__ OUT,         // [N,32]
                              int N)
{
    const int lane = threadIdx.x;
    const int m    = lane & 15;
    const int kh   = lane >> 4;
    const int h    = blockIdx.y;
    const int row0 = (blockIdx.x * blockDim.y + threadIdx.y) * 16;
    if (row0 >= N) return;  // wave-uniform; tiles are exact (N % 16 == 0)

    v8f acc = {};
    const float* Wh = WREL + h * 256;
    const float* ap = IN + (size_t)(row0 + m) * ldin + colOff + h * 16 + 2 * kh;
#pragma unroll
    for (int k = 0; k < 16; k += 4) {
        v2f av; av.x = ap[k]; av.y = ap[k + 1];
        const float* wp = Wh + (k + 2 * kh) * 16 + m;
        v2f bv; bv.x = wp[0]; bv.y = wp[16];
        acc = __builtin_amdgcn_wmma_f32_16x16x4_f32(
            false, av, false, bv, (short)0, acc, false, false);
    }
#pragma unroll
    for (int j = 0; j < 8; ++j)
        OUT[(size_t)(row0 + j + 8 * kh) * 32 + h * 16 + m] = acc[j];
}

// Pass 1: score per (edge, head) + atomic ordered-max per (dst-node, head)
__global__ void edge_scores(const int* __restrict__ ei, int E,
                            const float* __restrict__ kqv_dst,  // q at col 32, ld 96
                            const float* __restrict__ krel,     // [Nsrc,32]
                            const float* __restrict__ p_rel,    // [H]
                            float* __restrict__ scores,         // [E,H]
                            unsigned* __restrict__ amaxKey,     // [Ntot,H]
                            int segOff)
{
    const int e = blockIdx.x * blockDim.x + threadIdx.x;
    if (e >= E) return;
    const int src = ei[e];
    const int dst = ei[E + e];
    const float* q  = kqv_dst + (size_t)dst * 96 + 32;
    const float* kk = krel + (size_t)src * 32;
#pragma unroll
    for (int h = 0; h < HGT_H; ++h) {
        const float4* qv = (const float4*)(q + h * 16);
        const float4* kv = (const float4*)(kk + h * 16);
        float dot = 0.0f;
#pragma unroll
        for (int i = 0; i < 4; ++i) {
            float4 a = qv[i], b = kv[i];
            dot += a.x * b.x + a.y * b.y + a.z * b.z + a.w * b.w;
        }
        float s = dot * p_rel[h] * 0.25f;  // 1/sqrt(D), D=16
        scores[(size_t)e * HGT_H + h] = s;
        atomicMax(&amaxKey[(size_t)(segOff + dst) * HGT_H + h], f2key(s));
    }
}

// Pass 2: ex = exp(a - amax); denom += ex; acc[dst] += ex * v_rel[src]
__global__ void edge_aggregate(const int* __restrict__ ei, int E,
                               const float* __restrict__ scores,
                               const unsigned* __restrict__ amaxKey,
                               const float* __restrict__ vrel,   // [Nsrc,32]
                               float* __restrict__ denom,        // [Ntot,H]
                               float* __restrict__ acc,          // [Ntot,32]
                               int segOff)
{
    const int e = blockIdx.x * blockDim.x + threadIdx.x;
    if (e >= E) return;
    const int src = ei[e];
    const int seg = segOff + ei[E + e];
#pragma unroll
    for (int h = 0; h < HGT_H; ++h) {
        const float amax = key2f(amaxKey[(size_t)seg * HGT_H + h]);
        const float ex = expf(scores[(size_t)e * HGT_H + h] - amax);
        atomicAdd(&denom[(size_t)seg * HGT_H + h], ex);
        const float4* vv = (const float4*)(vrel + (size_t)src * 32 + h * 16);
        float* ap = acc + (size_t)seg * 32 + h * 16;
#pragma unroll
        for (int i = 0; i < 4; ++i) {
            float4 v = vv[i];
            atomicAdd(ap + 4 * i + 0, ex * v.x);
            atomicAdd(ap + 4 * i + 1, ex * v.y);
            atomicAdd(ap + 4 * i + 2, ex * v.z);
            atomicAdd(ap + 4 * i + 3, ex * v.w);
        }
    }
}

// Input order: x_paper, x_author, ei_writes, ei_rev_writes, ei_cites, ei_rev_cites,
// then params flattened with JAX-style sorted dict keys at every level:
//   conv1 (22 leaves at d_in[6]), conv2 (22 at d_in[28]), proj (4 at d_in[50]).
// Conv leaf layout (base B): a_rel[4 sorted names], kqv author{b,w}, paper{b,w},
//   m_rel[4], out author{b,w}, paper{b,w}, p_rel[4], skip author, skip paper.
extern "C" void kernel_launch(void* const* d_in, const int* in_sizes, int n_in,
                              void* d_out, int out_size, void* d_ws, size_t ws_size,
                              hipStream_t stream)
{
    (void)n_in; (void)out_size; (void)ws_size;
    // node types: 0 = paper, 1 = author (matches output concat order)
    const float* x[2] = { (const float*)d_in[0], (const float*)d_in[1] };
    const int* ei[4]  = { (const int*)d_in[2], (const int*)d_in[3],
                          (const int*)d_in[4], (const int*)d_in[5] };
    const int N[2]   = { in_sizes[0] / 256, in_sizes[1] / 256 };
    const int off[2] = { 0, N[0] };
    const int Ntot   = N[0] + N[1];
    const int E[4]   = { in_sizes[2] / 2, in_sizes[3] / 2, in_sizes[4] / 2, in_sizes[5] / 2 };

    // relations in processing order: writes, rev_writes, cites, rev_cites
    const int srcNT[4] = { 1, 0, 0, 0 };
    const int dstNT[4] = { 0, 1, 0, 0 };
    const int srt[4]   = { 0, 3, 1, 2 };  // index into sorted relation-name param arrays

    auto F = [&](int i) { return (const float*)d_in[i]; };
    const int C1 = 6, C2 = 28, PR = 50;
    const float* proj_b[2] = { F(PR + 2), F(PR + 0) };  // paper, author
    const float* proj_w[2] = { F(PR + 3), F(PR + 1) };

    // ---- workspace layout (256B aligned slices) ----
    size_t wo = 0;
    auto alloc = [&](size_t bytes) {
        void* p = (void*)((char*)d_ws + wo);
        wo = (wo + bytes + 255) & ~(size_t)255;
        return p;
    };
    float *h1[2], *h2[2], *kqvB[2], *krelB[4], *vrelB[4], *scB[4];
    for (int t = 0; t < 2; ++t) h1[t]   = (float*)alloc((size_t)N[t] * 32 * 4);
    for (int t = 0; t < 2; ++t) h2[t]   = (float*)alloc((size_t)N[t] * 32 * 4);
    for (int t = 0; t < 2; ++t) kqvB[t] = (float*)alloc((size_t)N[t] * 96 * 4);
    for (int r = 0; r < 4; ++r) krelB[r] = (float*)alloc((size_t)N[srcNT[r]] * 32 * 4);
    for (int r = 0; r < 4; ++r) vrelB[r] = (float*)alloc((size_t)N[srcNT[r]] * 32 * 4);
    for (int r = 0; r < 4; ++r) scB[r]   = (float*)alloc((size_t)E[r] * 2 * 4);
    unsigned* amaxK = (unsigned*)alloc((size_t)Ntot * 2 * 4);
    float* denomB   = (float*)alloc((size_t)Ntot * 2 * 4);
    float* accB     = (float*)alloc((size_t)Ntot * 32 * 4);

    const dim3 TB(32, 4);  // 4 wave32's per block
    // grid helper: rows covered per block = 64 * MT
    auto gx = [](int n, int mt) { return (unsigned)((n + 64 * mt - 1) / (64 * mt)); };

    // ---- stage 1: input projection 256->32 + ReLU ----
    for (int t = 0; t < 2; ++t)
        gemm_bias_act<2, 2><<<dim3(gx(N[t], 2)), TB, 0, stream>>>(
            x[t], 256, proj_w[t], proj_b[t], h1[t], 32, N[t], 256,
            nullptr, nullptr, 0, nullptr, 1);

    float* outF[2] = { (float*)d_out, (float*)d_out + (size_t)N[0] * 32 };

    for (int cv = 0; cv < 2; ++cv) {
        const int B = (cv == 0) ? C1 : C2;
        const float* hin[2] = { (cv == 0) ? h1[0] : h2[0], (cv == 0) ? h1[1] : h2[1] };
        float* hout[2]      = { (cv == 0) ? h2[0] : outF[0], (cv == 0) ? h2[1] : outF[1] };
        const int reluOut = (cv == 0) ? 1 : 0;

        const float *a_rel[4], *m_rel[4], *p_rel[4];
        for (int i = 0; i < 4; ++i) {
            a_rel[i] = F(B + i); m_rel[i] = F(B + 8 + i); p_rel[i] = F(B + 16 + i);
        }
        const float* kqv_b[2] = { F(B + 6),  F(B + 4)  };  // paper, author
        const float* kqv_w[2] = { F(B + 7),  F(B + 5)  };
        const float* out_b[2] = { F(B + 14), F(B + 12) };
        const float* out_w[2] = { F(B + 15), F(B + 13) };
        const float* skipG[2] = { F(B + 21), F(B + 20) };

        // kqv projection 32 -> 96
        for (int t = 0; t < 2; ++t)
            gemm_bias_act<6, 1><<<dim3(gx(N[t], 1)), TB, 0, stream>>>(
                hin[t], 32, kqv_w[t], kqv_b[t], kqvB[t], 96, N[t], 32,
                nullptr, nullptr, 0, nullptr, 0);

        // per-relation per-head K/V transforms (k part col 0, v part col 64)
        for (int r = 0; r < 4; ++r) {
            const int s = srcNT[r];
            rel_transform<<<dim3(gx(N[s], 1), 2), TB, 0, stream>>>(
                kqvB[s], 96, 0,  a_rel[srt[r]], krelB[r], N[s]);
            rel_transform<<<dim3(gx(N[s], 1), 2), TB, 0, stream>>>(
                kqvB[s], 96, 64, m_rel[srt[r]], vrelB[r], N[s]);
        }

        (void)hipMemsetAsync(amaxK,  0, (size_t)Ntot * 2 * 4, stream);
        (void)hipMemsetAsync(denomB, 0, (size_t)Ntot * 2 * 4, stream);
        (void)hipMemsetAsync(accB,   0, (size_t)Ntot * 32 * 4, stream);

        for (int r = 0; r < 4; ++r)
            edge_scores<<<dim3((E[r] + 255) / 256), dim3(256), 0, stream>>>(
                ei[r], E[r], kqvB[dstNT[r]], krelB[r], p_rel[srt[r]],
                scB[r], amaxK, off[dstNT[r]]);

        for (int r = 0; r < 4; ++r)
            edge_aggregate<<<dim3((E[r] + 255) / 256), dim3(256), 0, stream>>>(
                ei[r], E[r], scB[r], amaxK, vrelB[r], denomB, accB, off[dstNT[r]]);

        // normalize + GELU (fused into A-load) -> out linear -> gated skip (+ReLU conv1)
        for (int t = 0; t < 2; ++t)
            gemm_bias_act<2, 2><<<dim3(gx(N[t], 2)), TB, 0, stream>>>(
                accB + (size_t)off[t] * 32, 32, out_w[t], out_b[t], hout[t], 32, N[t], 32,
                denomB + (size_t)off[t] * 2, hin[t], 32, skipG[t], reluOut);
    }
}